// genpixel_80788334837911
// MI455X (gfx1250) — compile-verified
//
#include <hip/hip_runtime.h>
#include <hip/hip_bf16.h>

// Problem constants (match setup_inputs: B=16, C=2, H=W=256, CF=128, k=2, size=32)
#define BATCH   16
#define HW      65536      // 256*256
#define W_DIM   256
#define CF      128
#define NSLICE  32         // slices per batch in stage 1
#define SLICE   2048       // pixels per slice (HW / NSLICE)

// Output layout (flat concat, return order):
//   classiferT  [64,2]   @ 0      (128)
//   patchFeatDA [64,128] @ 128    (8192)
//   labelTTrue  [64,1]   @ 8320   (64)
//   labelpesudo [64,1]   @ 8384   (64)
//   provalue    [64,1]   @ 8448   (64)
//   pointXY     [64,2,2] @ 8512   (256)   total = 8768

// Monotone float -> uint32 key: a > b (float) <=> key(a) > key(b) (uint)
__device__ __forceinline__ unsigned fkey(float f) {
    unsigned u = __float_as_uint(f);
    return (u & 0x80000000u) ? ~u : (u | 0x80000000u);
}

// Merge two sorted top-2 lists (m1>=m2, o1>=o2) -> top-2 of the union into (m1,m2)
__device__ __forceinline__ void merge2(unsigned long long& m1, unsigned long long& m2,
                                       unsigned long long o1, unsigned long long o2) {
    if (o1 > m1) { m2 = (m1 > o2) ? m1 : o2; m1 = o1; }
    else         { m2 = (m2 > o1) ? m2 : o1; }
}

// ---------------------------------------------------------------------------
// Stage 1: one block per (batch, slice). Async-stage both logit planes of the
// slice into LDS (CDNA5 GLOBAL_LOAD_ASYNC_TO_LDS path, tracked by ASYNCcnt),
// compute d = x0 - x1, keep per-thread top-2 keys for +d (class 0) and -d
// (class 1), then LDS tree-reduce the (top1,top2) pairs. Thread 0 writes the
// block's 4 candidate keys to workspace.
// ---------------------------------------------------------------------------
__global__ __launch_bounds__(256) void topk_stage1(const float* __restrict__ infeat,
                                                   unsigned long long* __restrict__ ws) {
    const int b   = blockIdx.x >> 5;   // /NSLICE
    const int s   = blockIdx.x & 31;   // %NSLICE
    const int tid = threadIdx.x;

    __shared__ __align__(16) float smem[2 * SLICE];     // 16 KB: [0..2047]=x0, [2048..4095]=x1
    __shared__ unsigned long long red[4 * 256];         // 8 KB reduction scratch

    // Global byte addresses of this slice within plane0 / plane1 of batch b.
    const unsigned long long plane0 =
        (unsigned long long)(const char*)(infeat + ((size_t)b * 2) * HW + (size_t)s * SLICE);
    const unsigned long long plane1 = plane0 + (unsigned long long)HW * 4ull;

    // Low 32 bits of a generic LDS pointer are the wave-relative LDS byte offset.
    const unsigned lbase = (unsigned)(unsigned long long)(void*)smem;

    // 4 async B128 issues: 256 lanes x 16 B = 4 KB each; 2 chunks per 8 KB plane.
    #pragma unroll
    for (int q = 0; q < 2; ++q) {
        #pragma unroll
        for (int c = 0; c < 2; ++c) {
            unsigned           ldsa = lbase + (unsigned)(q * 8192 + c * 4096 + tid * 16);
            unsigned long long ga   = (q ? plane1 : plane0) + (unsigned long long)(c * 4096 + tid * 16);
            asm volatile("global_load_async_to_lds_b128 %0, %1, off"
                         :: "v"(ldsa), "v"(ga) : "memory");
        }
    }
    asm volatile("s_wait_asynccnt 0" ::: "memory");   // this wave's async ops done
    __syncthreads();                                   // all waves' LDS data visible

    // Per-thread top-2 keys over 8 strided pixels of the slice.
    unsigned long long a1 = 0, a2 = 0;   // class 0: top-2 of +d
    unsigned long long c1 = 0, c2 = 0;   // class 1: top-2 of -d
    const unsigned pbase = (unsigned)(s * SLICE);
    #pragma unroll
    for (int w = 0; w < 8; ++w) {
        int   l = tid + w * 256;
        float d = smem[l] - smem[SLICE + l];
        unsigned p = pbase + (unsigned)l;
        unsigned long long k0 = ((unsigned long long)fkey(d)  << 32) | (unsigned)(~p);
        unsigned long long k1 = ((unsigned long long)fkey(-d) << 32) | (unsigned)(~p);
        if (k0 > a1) { a2 = a1; a1 = k0; } else if (k0 > a2) { a2 = k0; }
        if (k1 > c1) { c2 = c1; c1 = k1; } else if (k1 > c2) { c2 = k1; }
    }

    red[tid] = a1; red[256 + tid] = a2; red[512 + tid] = c1; red[768 + tid] = c2;
    __syncthreads();
    for (int st = 128; st > 0; st >>= 1) {
        if (tid < st) {
            merge2(red[tid],       red[256 + tid], red[tid + st],       red[256 + tid + st]);
            merge2(red[512 + tid], red[768 + tid], red[512 + tid + st], red[768 + tid + st]);
        }
        __syncthreads();
    }
    if (tid == 0) {
        unsigned long long* o = ws + (size_t)blockIdx.x * 4;
        o[0] = red[0]; o[1] = red[256]; o[2] = red[512]; o[3] = red[768];
    }
}

// ---------------------------------------------------------------------------
// Stage 2: one block per batch. Merge the 32 per-slice candidate pairs per
// class, decode (pixel, d), then gather all outputs.
// ---------------------------------------------------------------------------
__global__ __launch_bounds__(128) void gather_stage2(const float* __restrict__ infeat,
                                                     const float* __restrict__ labelTpesudo,
                                                     const float* __restrict__ labelT,
                                                     const float* __restrict__ FeatureDA,
                                                     const unsigned long long* __restrict__ ws,
                                                     float* __restrict__ out) {
    const int b   = blockIdx.x;
    const int tid = threadIdx.x;

    __shared__ unsigned sp[4];    // pixel index per candidate (c*2 + j)
    __shared__ float    sdec[4];  // signed logit diff for that class (so prob = sigmoid(dec))

    if (tid == 0) {
        unsigned long long a1 = 0, a2 = 0, c1 = 0, c2 = 0;
        for (int s = 0; s < NSLICE; ++s) {
            const unsigned long long* o = ws + ((size_t)(b * NSLICE + s)) * 4;
            merge2(a1, a2, o[0], o[1]);
            merge2(c1, c2, o[2], o[3]);
        }
        unsigned long long keys[4] = { a1, a2, c1, c2 };
        #pragma unroll
        for (int m = 0; m < 4; ++m) {
            unsigned fk = (unsigned)(keys[m] >> 32);
            unsigned u  = (fk >> 31) ? (fk & 0x7FFFFFFFu) : ~fk;   // invert fkey
            sdec[m] = __uint_as_float(u);
            sp[m]   = ~((unsigned)keys[m]);                         // invert ~p
        }
    }
    __syncthreads();

    // patchFeatDA: 128 channels gathered by 128 threads per candidate.
    #pragma unroll
    for (int m = 0; m < 4; ++m) {
        const int cc = m >> 1, j = m & 1;
        const int r  = cc * 32 + b * 2 + j;          // row in concatenated outputs
        const unsigned p = sp[m];
        out[128 + r * CF + tid] = FeatureDA[((size_t)(b * CF + tid)) * HW + p];
    }

    if (tid < 4) {
        const int m  = tid;
        const int cc = m >> 1, j = m & 1;
        const int r  = cc * 32 + b * 2 + j;
        const unsigned p  = sp[m];
        const unsigned py = p >> 8;
        const unsigned px = p & (W_DIM - 1);

        // classiferT: raw logits at the point
        out[r * 2 + 0] = infeat[((size_t)(b * 2 + 0)) * HW + p];
        out[r * 2 + 1] = infeat[((size_t)(b * 2 + 1)) * HW + p];
        // labels
        out[8320 + r] = labelT[(size_t)b * HW + p];
        out[8384 + r] = labelTpesudo[(size_t)b * HW + p];
        // softmax probability of the selected class = sigmoid(dec)
        out[8448 + r] = 1.0f / (1.0f + __expf(-sdec[m]));
        // pointXY: [[px, min(px+31,255)], [py, min(py+31,255)]]
        unsigned pxe = px + 31u; if (pxe > 255u) pxe = 255u;
        unsigned pye = py + 31u; if (pye > 255u) pye = 255u;
        out[8512 + r * 4 + 0] = (float)px;
        out[8512 + r * 4 + 1] = (float)pxe;
        out[8512 + r * 4 + 2] = (float)py;
        out[8512 + r * 4 + 3] = (float)pye;
    }
}

extern "C" void kernel_launch(void* const* d_in, const int* in_sizes, int n_in,
                              void* d_out, int out_size, void* d_ws, size_t ws_size,
                              hipStream_t stream) {
    (void)in_sizes; (void)n_in; (void)out_size; (void)ws_size;
    const float* infeat       = (const float*)d_in[0];
    const float* labelTpesudo = (const float*)d_in[1];
    const float* labelT       = (const float*)d_in[2];
    const float* FeatureDA    = (const float*)d_in[3];
    unsigned long long* ws    = (unsigned long long*)d_ws;   // 512*4*8 B = 16 KB used
    float* out                = (float*)d_out;

    topk_stage1<<<BATCH * NSLICE, 256, 0, stream>>>(infeat, ws);
    gather_stage2<<<BATCH, 128, 0, stream>>>(infeat, labelTpesudo, labelT, FeatureDA, ws, out);
}